// LDAMLoss_1013612281881
// MI455X (gfx1250) — compile-verified
//
#include <hip/hip_runtime.h>
#include <math.h>

// LDAM loss: mean_i [ logsumexp_j(s*(x_ij - m_t*[j==t_i])) - s*(x_it - m_t) ]
// N=8192 rows, C=10000 cols, fp32. Memory-bound: one 327.7MB pass of HBM
// (~14.1us floor at 23.3 TB/s). No GEMM content -> WMMA inapplicable; the
// gfx1250 path that matters is async global->LDS staging + s_wait_asynccnt.
// Strategy: 1 block per row; async-DMA the row into LDS, apply margin in LDS,
// then 2-pass LSE (max, sum-exp2) entirely out of LDS. Deterministic mean.

#define TPB   256
#define MAX_C 10000

typedef int v4i __attribute__((ext_vector_type(4)));
typedef __attribute__((address_space(1))) v4i* glb_v4i_ptr;
typedef __attribute__((address_space(3))) v4i* lds_v4i_ptr;

__device__ __forceinline__ float warp_reduce_max(float v) {
#pragma unroll
  for (int o = 16; o > 0; o >>= 1) v = fmaxf(v, __shfl_xor(v, o, 32));
  return v;
}
__device__ __forceinline__ float warp_reduce_sum(float v) {
#pragma unroll
  for (int o = 16; o > 0; o >>= 1) v += __shfl_xor(v, o, 32);
  return v;
}

__global__ __launch_bounds__(TPB) void ldam_row_lse(
    const float* __restrict__ x, const int* __restrict__ target,
    const float* __restrict__ m_list, const int* __restrict__ s_ptr,
    float* __restrict__ row_loss, int c) {
  __shared__ __align__(16) float srow[MAX_C];
  __shared__ float sred[TPB / 32];
  __shared__ float s_bval[2];  // [0]: row max (raw x units), [1]: x_t - m_t

  const int row = blockIdx.x;
  const int tid = threadIdx.x;
  const float* rowp = x + (size_t)row * (size_t)c;
  const int nch = c >> 2;  // float4 chunks (C=10000 -> 2500, exact)

  // ---- Stage row into LDS via gfx1250 async global->LDS DMA ----
#if __has_builtin(__builtin_amdgcn_global_load_async_to_lds_b128)
  for (int i = tid; i < nch; i += TPB) {
    __builtin_amdgcn_global_load_async_to_lds_b128(
        (glb_v4i_ptr)(rowp + (size_t)i * 4),
        (lds_v4i_ptr)(srow + i * 4), /*offset=*/0, /*cpol=*/0);
  }
#if __has_builtin(__builtin_amdgcn_s_wait_asynccnt)
  __builtin_amdgcn_s_wait_asynccnt(0);
#else
  asm volatile("s_wait_asynccnt 0" ::: "memory");
#endif
#else
  for (int i = tid; i < nch; i += TPB)
    ((float4*)srow)[i] = ((const float4*)rowp)[i];
#endif
  // tail if c % 4 != 0 (zero iterations for C=10000)
  for (int i = (nch << 2) + tid; i < c; i += TPB) srow[i] = rowp[i];
  __syncthreads();

  const float sf = (float)s_ptr[0];  // s = 30

  // ---- Apply LDAM margin at the target column, in LDS ----
  if (tid == 0) {
    const int t = target[row];
    const float xv = srow[t] - m_list[t];
    srow[t] = xv;
    s_bval[1] = xv;
  }
  __syncthreads();

  const float4* s4 = (const float4*)srow;

  // ---- Pass 1: row max (raw x units; s > 0 so max commutes with scaling) ----
  float m = -3.402823466e+38f;
  for (int i = tid; i < nch; i += TPB) {
    float4 v = s4[i];
    m = fmaxf(m, fmaxf(fmaxf(v.x, v.y), fmaxf(v.z, v.w)));
  }
  for (int i = (nch << 2) + tid; i < c; i += TPB) m = fmaxf(m, srow[i]);
  m = warp_reduce_max(m);
  if ((tid & 31) == 0) sred[tid >> 5] = m;
  __syncthreads();
  if (tid == 0) {
    float mm = sred[0];
#pragma unroll
    for (int w = 1; w < TPB / 32; ++w) mm = fmaxf(mm, sred[w]);
    s_bval[0] = mm;
  }
  __syncthreads();
  const float M = sf * s_bval[0];  // max of scaled logits

  // ---- Pass 2: Z = sum exp(y - M), computed in exp2 domain ----
  const float K = 1.44269504088896340736f;  // log2(e)
  float z = 0.0f;
  for (int i = tid; i < nch; i += TPB) {
    float4 v = s4[i];
    z += __builtin_amdgcn_exp2f((sf * v.x - M) * K);
    z += __builtin_amdgcn_exp2f((sf * v.y - M) * K);
    z += __builtin_amdgcn_exp2f((sf * v.z - M) * K);
    z += __builtin_amdgcn_exp2f((sf * v.w - M) * K);
  }
  for (int i = (nch << 2) + tid; i < c; i += TPB)
    z += __builtin_amdgcn_exp2f((sf * srow[i] - M) * K);
  z = warp_reduce_sum(z);
  if ((tid & 31) == 0) sred[tid >> 5] = z;
  __syncthreads();
  if (tid == 0) {
    float Z = 0.0f;
#pragma unroll
    for (int w = 0; w < TPB / 32; ++w) Z += sred[w];
    const float LN2 = 0.69314718055994530942f;
    // loss_i = M + ln(Z) - y_t ;  ln(Z) = log2(Z)*ln2 via v_log_f32
    row_loss[row] = M + __builtin_amdgcn_logf(Z) * LN2 - sf * s_bval[1];
  }
}

__global__ __launch_bounds__(256) void ldam_mean(
    const float* __restrict__ row_loss, float* __restrict__ out, int n) {
  __shared__ float sred[8];
  float a = 0.0f;
  for (int i = threadIdx.x; i < n; i += 256) a += row_loss[i];
  a = warp_reduce_sum(a);
  if ((threadIdx.x & 31) == 0) sred[threadIdx.x >> 5] = a;
  __syncthreads();
  if (threadIdx.x == 0) {
    float t = 0.0f;
#pragma unroll
    for (int w = 0; w < 8; ++w) t += sred[w];
    out[0] = t / (float)n;
  }
}

extern "C" void kernel_launch(void* const* d_in, const int* in_sizes, int n_in,
                              void* d_out, int out_size, void* d_ws, size_t ws_size,
                              hipStream_t stream) {
  const float* x      = (const float*)d_in[0];
  const int*   target = (const int*)d_in[1];
  const float* m_list = (const float*)d_in[2];
  const int*   s_ptr  = (const int*)d_in[3];
  const int n = in_sizes[1];  // 8192
  const int c = in_sizes[2];  // 10000

  float* row_loss = (float*)d_ws;  // n floats of scratch

  ldam_row_lse<<<n, TPB, 0, stream>>>(x, target, m_list, s_ptr, row_loss, c);
  ldam_mean<<<1, 256, 0, stream>>>(row_loss, (float*)d_out, n);
}